// TaskHeads_76510547411303
// MI455X (gfx1250) — compile-verified
//
#include <hip/hip_runtime.h>

typedef __attribute__((ext_vector_type(2))) float v2f;
typedef __attribute__((ext_vector_type(8))) float v8f;

#define D_IN 128
#define H_HID 16
#define LDS_STRIDE 132   // 128 + 4 pad floats -> breaks 512B bank-conflict stride

// ---------------- bucketing kernels ----------------

__global__ void moe_init_kernel(int* __restrict__ counts, int* __restrict__ row_idx,
                                int E, int n_pad) {
    int i = blockIdx.x * blockDim.x + threadIdx.x;
    if (i < E) counts[i] = 0;
    if (i < n_pad) row_idx[i] = -1;
}

__global__ void moe_count_kernel(const int* __restrict__ sid, int* __restrict__ counts, int n) {
    int i = blockIdx.x * blockDim.x + threadIdx.x;
    if (i < n) atomicAdd(&counts[sid[i]], 1);
}

__global__ void moe_scan_kernel(const int* __restrict__ counts, int* __restrict__ tileStart,
                                int* __restrict__ cursors, int E) {
    if (blockIdx.x == 0 && threadIdx.x == 0) {
        int t = 0;
        for (int e = 0; e < E; ++e) {
            tileStart[e] = t;
            cursors[e]   = t * 16;           // padded row start of expert e
            t += (counts[e] + 15) >> 4;      // tiles per expert (16-row tiles)
        }
        tileStart[E] = t;                    // total tiles
    }
}

__global__ void moe_scatter_kernel(const int* __restrict__ sid, int* __restrict__ cursors,
                                   int* __restrict__ row_idx, int n) {
    int i = blockIdx.x * blockDim.x + threadIdx.x;
    if (i < n) {
        int pos = atomicAdd(&cursors[sid[i]], 1);
        row_idx[pos] = i;
    }
}

// ---------------- main WMMA MoE kernel: one wave per 16-row tile ----------------

__launch_bounds__(32)
__global__ void moe_mlp_kernel(const float* __restrict__ x,
                               const float* __restrict__ W1,
                               const float* __restrict__ b1,
                               const float* __restrict__ W2,
                               const float* __restrict__ b2,
                               const int* __restrict__ row_idx,
                               const int* __restrict__ tileStart,
                               float* __restrict__ out, int E) {
    __shared__ float ldsx[16 * LDS_STRIDE];

    const int t = blockIdx.x;
    const int total = tileStart[E];
    if (t >= total) return;                  // wave-uniform exit

    // which expert owns tile t (E <= 21, cheap linear scan, scalar)
    int e = 0;
    while (e + 1 < E && tileStart[e + 1] <= t) ++e;
    const int base = t * 16;                 // tiles are globally contiguous 16-row slots

    const int lane = threadIdx.x;            // wave32
    const int half = lane >> 4;              // 0: lanes 0-15, 1: lanes 16-31
    const int lid  = lane & 15;

    // Stage 16 gathered rows of x into LDS; each row is 512B contiguous,
    // the wave covers it with one coalesced b128 load (32 lanes * 16B).
    for (int r = 0; r < 16; ++r) {
        int row = row_idx[base + r];
        float4 v = make_float4(0.f, 0.f, 0.f, 0.f);
        if (row >= 0)
            v = ((const float4*)(x + (size_t)row * D_IN))[lane];
        *(float4*)(&ldsx[r * LDS_STRIDE + lane * 4]) = v;
    }
    __syncthreads();

    // h(16x16) = xTile(16x128) @ W1[e](128x16): 32 chained V_WMMA_F32_16X16X4_F32.
    // 32-bit A 16x4 layout: lanes 0-15 hold K=0,1; lanes 16-31 hold K=2,3 (M = lid).
    // B 4x16: VGPR0 = K{0|2}, VGPR1 = K{1|3} per lane half, N = lid.
    const float* w1e = W1 + (size_t)e * (D_IN * H_HID);
    v8f acc = {};
    #pragma unroll
    for (int k = 0; k < D_IN / 4; ++k) {
        const int kb = 4 * k + 2 * half;
        v2f a, b;
        a.x = ldsx[lid * LDS_STRIDE + kb + 0];   // ds_load_b64, conflict-free (pad)
        a.y = ldsx[lid * LDS_STRIDE + kb + 1];
        b.x = w1e[(kb + 0) * H_HID + lid];       // W1 tile hot in L1/L2
        b.y = w1e[(kb + 1) * H_HID + lid];
        acc = __builtin_amdgcn_wmma_f32_16x16x4_f32(
            false, a, false, b, (short)0, acc, false, false);
    }

    // C/D layout: VGPR v, lanes 0-15 -> M=v; lanes 16-31 -> M=v+8; N = lid.
    const float w2v = W2[e * H_HID + lid];
    const float b1v = b1[e * H_HID + lid];
    const float b2v = b2[e];
    #pragma unroll
    for (int v = 0; v < 8; ++v) {
        float h = fmaxf(acc[v] + b1v, 0.0f);     // bias + ReLU (layer 1)
        float p = h * w2v;
        // reduce over the 16 N-lanes of this half-wave
        p += __shfl_xor(p, 1, 32);
        p += __shfl_xor(p, 2, 32);
        p += __shfl_xor(p, 4, 32);
        p += __shfl_xor(p, 8, 32);
        if (lid == 0) {
            int m = v + 8 * half;
            int row = row_idx[base + m];
            if (row >= 0)
                out[row] = fmaxf(p + b2v, 0.0f); // layer 2 + ReLU, scatter
        }
    }
}

// ---------------- launcher ----------------

extern "C" void kernel_launch(void* const* d_in, const int* in_sizes, int n_in,
                              void* d_out, int out_size, void* d_ws, size_t ws_size,
                              hipStream_t stream) {
    const float* x  = (const float*)d_in[0];
    const int*   sid= (const int*)  d_in[1];
    const float* W1 = (const float*)d_in[2];
    const float* b1 = (const float*)d_in[3];
    const float* W2 = (const float*)d_in[4];
    const float* b2 = (const float*)d_in[5];
    float* out = (float*)d_out;

    const int N = in_sizes[1];          // rows (sid count)
    const int E = in_sizes[5];          // experts (b2 count)

    // workspace layout (ints): counts[E] @0, tileStart[E+1] @32, cursors[E] @64,
    // row_idx[N + 16E] @128
    int* ws        = (int*)d_ws;
    int* counts    = ws;
    int* tileStart = ws + 32;
    int* cursors   = ws + 64;
    int* row_idx   = ws + 128;

    const int n_pad = N + 16 * E;
    const int maxTiles = (N + 15) / 16 + E;

    moe_init_kernel   <<<(n_pad + 255) / 256, 256, 0, stream>>>(counts, row_idx, E, n_pad);
    moe_count_kernel  <<<(N + 255) / 256,     256, 0, stream>>>(sid, counts, N);
    moe_scan_kernel   <<<1, 1, 0, stream>>>(counts, tileStart, cursors, E);
    moe_scatter_kernel<<<(N + 255) / 256,     256, 0, stream>>>(sid, cursors, row_idx, N);
    moe_mlp_kernel    <<<maxTiles, 32, 0, stream>>>(x, W1, b1, W2, b2,
                                                    row_idx, tileStart, out, E);
}